// DoubleConv2_51986284151365
// MI455X (gfx1250) — compile-verified
//
#include <hip/hip_runtime.h>

typedef __attribute__((ext_vector_type(2))) float v2f;
typedef __attribute__((ext_vector_type(8))) float v8f;

#define B_    8
#define H_    128
#define W_    128
#define KK_   9
#define COUT_ 128
#define EPS_  1e-5f

// ---------------------------------------------------------------------------
// 3x3 offset conv (18 output channels). One block = 256 consecutive output
// pixels of a single (b, co) plane; weight row staged in LDS.
// ---------------------------------------------------------------------------
template <int CIN>
__global__ __launch_bounds__(256) void offset_conv_kernel(
    const float* __restrict__ x, const float* __restrict__ ow,
    const float* __restrict__ ob, float* __restrict__ off) {
  __shared__ float sw[CIN * 9];
  const int tid = threadIdx.x;
  const long gidx = (long)blockIdx.x * 256 + tid;
  const int w  = (int)(gidx % W_);
  const int h  = (int)((gidx / W_) % H_);
  const int co = (int)((gidx / ((long)H_ * W_)) % (2 * KK_));
  const int b  = (int)(gidx / ((long)2 * KK_ * H_ * W_));

  for (int i = tid; i < CIN * 9; i += 256) sw[i] = ow[co * CIN * 9 + i];
  __syncthreads();

  const float* xb = x + (long)b * CIN * H_ * W_;
  float acc = ob[co];
  for (int c = 0; c < CIN; ++c) {
    const float* xc = xb + (long)c * H_ * W_;
    const float* wc = sw + c * 9;
#pragma unroll
    for (int dy = 0; dy < 3; ++dy) {
      const int hh = h + dy - 1;
      if (hh < 0 || hh >= H_) continue;
#pragma unroll
      for (int dx = 0; dx < 3; ++dx) {
        const int ww = w + dx - 1;
        if (ww < 0 || ww >= W_) continue;
        acc += xc[hh * W_ + ww] * wc[dy * 3 + dx];
      }
    }
  }
  off[gidx] = acc;
}

// ---------------------------------------------------------------------------
// Deformable conv as tiled GEMM via V_WMMA_F32_16X16X4_F32.
// Block = 256 threads (8 wave32) handles 16 pixels x 128 out channels.
// ---------------------------------------------------------------------------
__device__ __forceinline__ float corner_val(const float* __restrict__ xc,
                                            int yi, int xi, float wgt) {
  const bool val = (yi >= 0) & (yi < H_) & (xi >= 0) & (xi < W_);
  const int yc = min(max(yi, 0), H_ - 1);
  const int xcl = min(max(xi, 0), W_ - 1);
  return val ? xc[yc * W_ + xcl] * wgt : 0.f;
}

template <int CIN>
__global__ __launch_bounds__(256) void deform_wmma_kernel(
    const float* __restrict__ x, const float* __restrict__ off,
    const float* __restrict__ dw, const float* __restrict__ db,
    float* __restrict__ y) {
  constexpr int K = CIN * KK_;           // 576 or 1152
  __shared__ float sA[K * 16];           // A panel, [k][m], m fastest
  __shared__ float s_wy[144], s_wx[144];
  __shared__ int   s_y0[144], s_x0[144];

  const int tid = threadIdx.x;
  const int w0 = blockIdx.x * 16;
  const int h  = blockIdx.y;
  const int b  = blockIdx.z;

  // Phase 0: bilinear coordinates for the 16 pixels x 9 taps
  if (tid < 144) {
    const int k = tid >> 4, m = tid & 15;
    const int w = w0 + m;
    const float oy = off[(((long)b * 18 + 2 * k + 0) * H_ + h) * W_ + w];
    const float ox = off[(((long)b * 18 + 2 * k + 1) * H_ + h) * W_ + w];
    const float ys = oy + (float)(k / 3 - 1) + (float)h;
    const float xs = ox + (float)(k % 3 - 1) + (float)w;
    const float y0f = floorf(ys), x0f = floorf(xs);
    s_wy[tid] = ys - y0f;
    s_wx[tid] = xs - x0f;
    s_y0[tid] = (int)y0f;
    s_x0[tid] = (int)x0f;
  }
  __syncthreads();

  // Phase 1: gather sampled A panel into LDS (all gathers are L2 hits)
  const float* xb = x + (long)b * CIN * H_ * W_;
  for (int idx = tid; idx < 16 * KK_ * CIN; idx += 256) {
    const int c = idx / 144;
    const int rem = idx % 144;           // k*16 + m
    const int k = rem >> 4, m = rem & 15;
    const int y0 = s_y0[rem], x0 = s_x0[rem];
    const float wy = s_wy[rem], wx = s_wx[rem];
    const float* xc = xb + (long)c * H_ * W_;
    float v = corner_val(xc, y0,     x0,     (1.f - wy) * (1.f - wx));
    v      += corner_val(xc, y0,     x0 + 1, (1.f - wy) * wx);
    v      += corner_val(xc, y0 + 1, x0,     wy * (1.f - wx));
    v      += corner_val(xc, y0 + 1, x0 + 1, wy * wx);
    sA[(c * KK_ + k) * 16 + m] = v;
  }
  __syncthreads();

  // Phase 2: K-loop of fp32 WMMA. Wave w owns out channels [16w, 16w+16).
  const int lane = tid & 31;
  const int wave = tid >> 5;
  const int n  = lane & 15;              // N column (and A row M)
  const int kh = lane >> 4;              // K half select
  const int o  = wave * 16 + n;          // output channel
  const float* brow = dw + (long)o * K;  // weight row, contiguous in k

  v8f acc = {};
  for (int s = 0; s < K / 4; ++s) {
    const int kk = 4 * s + 2 * kh;
    v2f a;
    a.x = sA[kk * 16 + n];               // (m=lane&15, K=kk)
    a.y = sA[(kk + 1) * 16 + n];         // (m, K=kk+1)
    const v2f bb = *(const v2f*)(brow + kk);  // b64, 8B aligned (K even, kk even)
    acc = __builtin_amdgcn_wmma_f32_16x16x4_f32(
        false, a, false, bb, (short)0, acc, false, false);
  }

  // Epilogue: C/D layout -> lane half selects M+8; add bias, scatter store.
  const float bias = db[o];
#pragma unroll
  for (int r = 0; r < 8; ++r) {
    const int M = r + kh * 8;
    y[(((long)b * COUT_ + o) * H_ + h) * W_ + (w0 + M)] = acc[r] + bias;
  }
}

// ---------------------------------------------------------------------------
// BatchNorm: deterministic two-stage per-channel mean/var reduction
// ---------------------------------------------------------------------------
__global__ __launch_bounds__(256) void bn_stats1_kernel(
    const float* __restrict__ yin, float* __restrict__ partials) {
  __shared__ float ss[256], sq[256];
  const int c = blockIdx.y;
  const int slab = blockIdx.x;           // 16 slabs of 8192 elements
  const int tid = threadIdx.x;
  float s = 0.f, q = 0.f;
  for (int j = 0; j < 32; ++j) {
    const int i = slab * 8192 + j * 256 + tid;   // i in [0, 131072)
    const int bb = i >> 14;              // / (H*W)
    const int pos = i & 16383;
    const float v = yin[((long)bb * COUT_ + c) * H_ * W_ + pos];
    s += v;
    q += v * v;
  }
  ss[tid] = s; sq[tid] = q;
  __syncthreads();
  for (int st = 128; st > 0; st >>= 1) {
    if (tid < st) { ss[tid] += ss[tid + st]; sq[tid] += sq[tid + st]; }
    __syncthreads();
  }
  if (tid == 0) {
    partials[(c * 16 + slab) * 2 + 0] = ss[0];
    partials[(c * 16 + slab) * 2 + 1] = sq[0];
  }
}

__global__ void bn_stats2_kernel(const float* __restrict__ partials,
                                 float* __restrict__ stats) {
  const int c = threadIdx.x;
  if (c < COUT_) {
    float s = 0.f, q = 0.f;
    for (int i = 0; i < 16; ++i) {
      s += partials[(c * 16 + i) * 2 + 0];
      q += partials[(c * 16 + i) * 2 + 1];
    }
    const float N = (float)((long)B_ * H_ * W_);
    const float mean = s / N;
    const float var = q / N - mean * mean;
    stats[c] = mean;
    stats[COUT_ + c] = rsqrtf(var + EPS_);
  }
}

__global__ __launch_bounds__(256) void bn_apply_kernel(
    const float* __restrict__ yin, const float* __restrict__ stats,
    const float* __restrict__ gamma, const float* __restrict__ beta,
    float* __restrict__ out) {
  const long idx = (long)blockIdx.x * 256 + threadIdx.x;
  const int c = (int)((idx >> 14) & (COUT_ - 1));
  const float v = yin[idx];
  const float r = (v - stats[c]) * stats[COUT_ + c] * gamma[c] + beta[c];
  out[idx] = fmaxf(r, 0.f);
}

// ---------------------------------------------------------------------------
extern "C" void kernel_launch(void* const* d_in, const int* in_sizes, int n_in,
                              void* d_out, int out_size, void* d_ws, size_t ws_size,
                              hipStream_t stream) {
  (void)in_sizes; (void)n_in; (void)out_size; (void)ws_size;
  const float* x   = (const float*)d_in[0];
  const float* ow1 = (const float*)d_in[1];
  const float* ob1 = (const float*)d_in[2];
  const float* dw1 = (const float*)d_in[3];
  const float* db1 = (const float*)d_in[4];
  const float* g1  = (const float*)d_in[5];
  const float* b1  = (const float*)d_in[6];
  const float* ow2 = (const float*)d_in[7];
  const float* ob2 = (const float*)d_in[8];
  const float* dw2 = (const float*)d_in[9];
  const float* db2 = (const float*)d_in[10];
  const float* g2  = (const float*)d_in[11];
  const float* b2  = (const float*)d_in[12];

  float* ws = (float*)d_ws;
  const long OFF_SZ = (long)B_ * 18 * H_ * W_;       // 2,359,296
  const long FEA_SZ = (long)B_ * COUT_ * H_ * W_;    // 16,777,216
  float* off   = ws;
  float* bufB  = off + OFF_SZ;
  float* bufC  = bufB + FEA_SZ;
  float* parts = bufC + FEA_SZ;
  float* stats = parts + 128 * 16 * 2;

  const dim3 blk(256);
  const dim3 dgrid(W_ / 16, H_, B_);

  // ---- layer 1 ----
  offset_conv_kernel<64><<<(int)(OFF_SZ / 256), blk, 0, stream>>>(x, ow1, ob1, off);
  deform_wmma_kernel<64><<<dgrid, blk, 0, stream>>>(x, off, dw1, db1, bufB);
  bn_stats1_kernel<<<dim3(16, 128), blk, 0, stream>>>(bufB, parts);
  bn_stats2_kernel<<<1, 128, 0, stream>>>(parts, stats);
  bn_apply_kernel<<<(int)(FEA_SZ / 256), blk, 0, stream>>>(bufB, stats, g1, b1, bufB);

  // ---- layer 2 ----
  offset_conv_kernel<128><<<(int)(OFF_SZ / 256), blk, 0, stream>>>(bufB, ow2, ob2, off);
  deform_wmma_kernel<128><<<dgrid, blk, 0, stream>>>(bufB, off, dw2, db2, bufC);
  bn_stats1_kernel<<<dim3(16, 128), blk, 0, stream>>>(bufC, parts);
  bn_stats2_kernel<<<1, 128, 0, stream>>>(parts, stats);
  bn_apply_kernel<<<(int)(FEA_SZ / 256), blk, 0, stream>>>(bufC, stats, g2, b2,
                                                           (float*)d_out);
}